// ComplementaryNodeEdgePrompt_34248069218340
// MI455X (gfx1250) — compile-verified
//
#include <hip/hip_runtime.h>
#include <math.h>

#define N_NODES 50000
#define N_EDGES 640000
#define DIM     128
#define NA      5

typedef float v2f __attribute__((ext_vector_type(2)));
typedef float v8f __attribute__((ext_vector_type(8)));

// ---------------------------------------------------------------- zero
__global__ void zero_kernel(float4* __restrict__ p, int n4) {
    int i = blockIdx.x * blockDim.x + threadIdx.x;
    int stride = gridDim.x * blockDim.x;
    float4 z = make_float4(0.f, 0.f, 0.f, 0.f);
    for (; i < n4; i += stride) p[i] = z;
}

// ---------------------------------------------------------------- node prompt
// node_w = softmax(x @ att_w + att_b); node_prompted = x + node_w @ node_anchor
__global__ __launch_bounds__(256) void node_kernel(
    const float* __restrict__ x, const float* __restrict__ node_anchor,
    const float* __restrict__ att_w, const float* __restrict__ att_b,
    float* __restrict__ out_np)
{
    __shared__ float s_attw[DIM * NA];    // [d][a]
    __shared__ float s_anchor[NA * DIM];  // [a][d]
    __shared__ float s_attb[NA];
    int tid = threadIdx.x;
    for (int i = tid; i < DIM * NA; i += 256) { s_attw[i] = att_w[i]; s_anchor[i] = node_anchor[i]; }
    if (tid < NA) s_attb[tid] = att_b[tid];
    __syncthreads();

    int n = blockIdx.x * 256 + tid;
    if (n >= N_NODES) return;
    const float4* xr4 = (const float4*)(x + (size_t)n * DIM);

    float acc[NA];
#pragma unroll
    for (int a = 0; a < NA; ++a) acc[a] = s_attb[a];
    for (int d4 = 0; d4 < DIM / 4; ++d4) {
        float4 xv = xr4[d4];
#pragma unroll
        for (int a = 0; a < NA; ++a) {
            acc[a] += xv.x * s_attw[(d4 * 4 + 0) * NA + a]
                    + xv.y * s_attw[(d4 * 4 + 1) * NA + a]
                    + xv.z * s_attw[(d4 * 4 + 2) * NA + a]
                    + xv.w * s_attw[(d4 * 4 + 3) * NA + a];
        }
    }
    float mx = acc[0];
#pragma unroll
    for (int a = 1; a < NA; ++a) mx = fmaxf(mx, acc[a]);
    float s = 0.0f;
#pragma unroll
    for (int a = 0; a < NA; ++a) { acc[a] = __expf(acc[a] - mx); s += acc[a]; }
    float inv = 1.0f / s;
#pragma unroll
    for (int a = 0; a < NA; ++a) acc[a] *= inv;

    float4* o4 = (float4*)(out_np + (size_t)n * DIM);
    for (int d4 = 0; d4 < DIM / 4; ++d4) {
        float4 v = xr4[d4];
#pragma unroll
        for (int a = 0; a < NA; ++a) {
            v.x += acc[a] * s_anchor[a * DIM + d4 * 4 + 0];
            v.y += acc[a] * s_anchor[a * DIM + d4 * 4 + 1];
            v.z += acc[a] * s_anchor[a * DIM + d4 * 4 + 2];
            v.w += acc[a] * s_anchor[a * DIM + d4 * 4 + 3];
        }
        o4[d4] = v;
    }
}

// ---------------------------------------------------------------- edge prompt (WMMA)
// One wave32 per 16-edge tile.
// Stage 1: logits[16x16] = combined[16x256] @ edge_w[256x16(pad)]   -> 64x v_wmma_f32_16x16x4_f32
// Stage 2: prompt[16x128] = b[16x8(pad)] @ edge_anchor[8(pad)x128]  -> 16x v_wmma_f32_16x16x4_f32
__global__ __launch_bounds__(256) void edge_kernel(
    const float* __restrict__ x,
    const float* __restrict__ edge_anchor,
    const float* __restrict__ edge_w,
    const float* __restrict__ edge_b,
    const int*   __restrict__ ei,           // [2*E]
    float* __restrict__ edge_prompt,        // [E*DIM]
    float* __restrict__ edge_agg)           // [N*DIM], zeroed, atomics
{
    __shared__ float s_ew16[2 * DIM * 16];  // edge_w padded to 16 cols (16 KB), cols>=5 zero
    __shared__ float s_ea[8 * DIM];         // edge_anchor padded to 8 rows (4 KB)
    __shared__ float s_eb[8];
    __shared__ float s_ltile[8][16 * 16];   // per-wave logit tile
    __shared__ float s_btile[8][16 * 8];    // per-wave softmax weights, K padded to 8
    __shared__ int   s_idx[8][32];          // [wave][0..15]=src, [16..31]=dst

    int tid = threadIdx.x;
    for (int i = tid; i < 2 * DIM * 16; i += 256) {
        int k = i >> 4, n = i & 15;
        s_ew16[i] = (n < NA) ? edge_w[k * NA + n] : 0.0f;
    }
    for (int i = tid; i < 8 * DIM; i += 256) s_ea[i] = (i < NA * DIM) ? edge_anchor[i] : 0.0f;
    if (tid < 8) s_eb[tid] = (tid < NA) ? edge_b[tid] : 0.0f;
    __syncthreads();

    const int wave = tid >> 5;
    const int lane = tid & 31;
    const int hh   = lane >> 4;        // half-wave select (K parity pairs)
    const int l16  = lane & 15;

    long base = ((long)blockIdx.x * 8 + wave) * 16;
    if (base >= N_EDGES) return;       // wave-uniform

    // non-divergent endpoint preload: lanes 0-15 -> src, 16-31 -> dst
    {
        long e = base + l16; if (e >= N_EDGES) e = N_EDGES - 1;
        s_idx[wave][lane] = ei[(size_t)hh * N_EDGES + e];
    }
    const float* pa_s = x + (size_t)s_idx[wave][l16] * DIM + 2 * hh;       // own edge, src row
    const float* pa_d = x + (size_t)s_idx[wave][16 + l16] * DIM + 2 * hh;  // own edge, dst row
    const float* bp   = s_ew16 + 32 * hh + l16;                            // B frag base (k0=0)

    // ---- stage 1: K = 0..127 (src rows), then 128..255 (dst rows), chunks of 4
    v8f c = {};
#pragma unroll 8
    for (int k0 = 0; k0 < 32; ++k0) {
        v2f a = *(const v2f*)(pa_s + k0 * 4);              // global_load_b64, imm offset
        v2f b = { bp[k0 * 64], bp[k0 * 64 + 16] };         // ds_load_b32, imm offsets
        c = __builtin_amdgcn_wmma_f32_16x16x4_f32(false, a, false, b, (short)0, c, false, false);
    }
#pragma unroll 8
    for (int k0 = 0; k0 < 32; ++k0) {
        v2f a = *(const v2f*)(pa_d + k0 * 4);
        v2f b = { bp[(k0 + 32) * 64], bp[(k0 + 32) * 64 + 16] };
        c = __builtin_amdgcn_wmma_f32_16x16x4_f32(false, a, false, b, (short)0, c, false, false);
    }

    // ---- scatter C tile to LDS:  ltile[m][n],  m = edge-in-tile, n = anchor
    float* lt = s_ltile[wave];
#pragma unroll
    for (int r = 0; r < 8; ++r) {
        int m = hh ? (r + 8) : r;
        lt[m * 16 + l16] = c[r];
    }

    // ---- leaky_relu + bias + softmax per edge (lanes 0..15)
    float* bt = s_btile[wave];
    if (lane < 16) {
        float lr[NA];
        float mx = -3.0e38f;
#pragma unroll
        for (int a = 0; a < NA; ++a) {
            float v = lt[lane * 16 + a] + s_eb[a];
            v = (v > 0.0f) ? v : 0.01f * v;        // jax leaky_relu slope 0.01
            lr[a] = v;
            mx = fmaxf(mx, v);
        }
        float s = 0.0f;
#pragma unroll
        for (int a = 0; a < NA; ++a) { lr[a] = __expf(lr[a] - mx); s += lr[a]; }
        float inv = 1.0f / s;
#pragma unroll
        for (int a = 0; a < NA; ++a) bt[lane * 8 + a] = lr[a] * inv;
#pragma unroll
        for (int a = NA; a < 8; ++a) bt[lane * 8 + a] = 0.0f;
    }

    // ---- stage 2 A fragments: b-tile rows, K padded to 8 (2 wmma k-steps)
    v2f a0 = *(const v2f*)(bt + l16 * 8 + 2 * hh);       // K = 0..3
    v2f a1 = *(const v2f*)(bt + l16 * 8 + 4 + 2 * hh);   // K = 4..7 (5..7 zero)

    const float* eap = s_ea + 2 * hh * DIM + l16;        // B frag base for stage 2
    const bool full = (base + 16 <= (long)N_EDGES);      // wave-uniform

#pragma unroll 2
    for (int ft = 0; ft < 8; ++ft) {
        int col = ft * 16 + l16;
        v2f b0 = { eap[ft * 16],            eap[DIM + ft * 16] };
        v2f b1 = { eap[4 * DIM + ft * 16],  eap[5 * DIM + ft * 16] };
        v8f d = {};
        d = __builtin_amdgcn_wmma_f32_16x16x4_f32(false, a0, false, b0, (short)0, d, false, false);
        d = __builtin_amdgcn_wmma_f32_16x16x4_f32(false, a1, false, b1, (short)0, d, false, false);

        if (full) {
#pragma unroll
            for (int r = 0; r < 8; ++r) {
                int m = hh ? (r + 8) : r;
                long e = base + m;
                float v = d[r];
                edge_prompt[e * DIM + col] = v;
                atomicAdd(edge_agg + (size_t)s_idx[wave][m] * DIM + col, v);
                atomicAdd(edge_agg + (size_t)s_idx[wave][16 + m] * DIM + col, v);
            }
        } else {
#pragma unroll
            for (int r = 0; r < 8; ++r) {
                int m = hh ? (r + 8) : r;
                long e = base + m;
                if (e < N_EDGES) {
                    float v = d[r];
                    edge_prompt[e * DIM + col] = v;
                    atomicAdd(edge_agg + (size_t)s_idx[wave][m] * DIM + col, v);
                    atomicAdd(edge_agg + (size_t)s_idx[wave][16 + m] * DIM + col, v);
                }
            }
        }
    }
}

// ---------------------------------------------------------------- gated fusion
__global__ __launch_bounds__(256) void gate_kernel(
    const float* __restrict__ np, const float* __restrict__ ea,
    const float* __restrict__ gate_w, const float* __restrict__ gate_b,
    float* __restrict__ out_final)
{
    __shared__ float s_gw[2 * DIM * 2];   // [k][j]
    __shared__ float s_gb[2];
    int tid = threadIdx.x;
    for (int i = tid; i < 2 * DIM * 2; i += 256) s_gw[i] = gate_w[i];
    if (tid < 2) s_gb[tid] = gate_b[tid];
    __syncthreads();

    int n = blockIdx.x * 256 + tid;
    if (n >= N_NODES) return;
    const float4* pr4 = (const float4*)(np + (size_t)n * DIM);
    const float4* er4 = (const float4*)(ea + (size_t)n * DIM);

    float g0 = s_gb[0], g1 = s_gb[1];
    for (int d4 = 0; d4 < DIM / 4; ++d4) {
        float4 a = pr4[d4];
        float4 b = er4[d4];
        int d = d4 * 4;
        g0 += a.x * s_gw[(d + 0) * 2] + a.y * s_gw[(d + 1) * 2]
            + a.z * s_gw[(d + 2) * 2] + a.w * s_gw[(d + 3) * 2]
            + b.x * s_gw[(DIM + d + 0) * 2] + b.y * s_gw[(DIM + d + 1) * 2]
            + b.z * s_gw[(DIM + d + 2) * 2] + b.w * s_gw[(DIM + d + 3) * 2];
        g1 += a.x * s_gw[(d + 0) * 2 + 1] + a.y * s_gw[(d + 1) * 2 + 1]
            + a.z * s_gw[(d + 2) * 2 + 1] + a.w * s_gw[(d + 3) * 2 + 1]
            + b.x * s_gw[(DIM + d + 0) * 2 + 1] + b.y * s_gw[(DIM + d + 1) * 2 + 1]
            + b.z * s_gw[(DIM + d + 2) * 2 + 1] + b.w * s_gw[(DIM + d + 3) * 2 + 1];
    }
    float mx = fmaxf(g0, g1);
    float e0 = __expf(g0 - mx), e1 = __expf(g1 - mx);
    float inv = 1.0f / (e0 + e1);
    e0 *= inv; e1 *= inv;

    float4* o4 = (float4*)(out_final + (size_t)n * DIM);
    for (int d4 = 0; d4 < DIM / 4; ++d4) {
        float4 a = pr4[d4];
        float4 b = er4[d4];
        float4 v;
        v.x = e0 * a.x + e1 * b.x;
        v.y = e0 * a.y + e1 * b.y;
        v.z = e0 * a.z + e1 * b.z;
        v.w = e0 * a.w + e1 * b.w;
        o4[d4] = v;
    }
}

// ---------------------------------------------------------------- launcher
extern "C" void kernel_launch(void* const* d_in, const int* in_sizes, int n_in,
                              void* d_out, int out_size, void* d_ws, size_t ws_size,
                              hipStream_t stream) {
    const float* x           = (const float*)d_in[0];
    const float* node_anchor = (const float*)d_in[1];
    const float* att_w       = (const float*)d_in[2];
    const float* att_b       = (const float*)d_in[3];
    const float* edge_anchor = (const float*)d_in[4];
    const float* edge_w      = (const float*)d_in[5];
    const float* edge_b      = (const float*)d_in[6];
    const float* gate_w      = (const float*)d_in[7];
    const float* gate_b      = (const float*)d_in[8];
    const int*   edge_index  = (const int*)d_in[9];

    float* out           = (float*)d_out;
    float* final_x       = out;                                  // [N*D]
    float* edge_prompt   = out + (size_t)N_NODES * DIM;          // [E*D]
    float* node_prompted = edge_prompt + (size_t)N_EDGES * DIM;  // [N*D]
    float* edge_agg      = node_prompted + (size_t)N_NODES * DIM;// [N*D]

    zero_kernel<<<1600, 256, 0, stream>>>((float4*)edge_agg, N_NODES * DIM / 4);
    node_kernel<<<(N_NODES + 255) / 256, 256, 0, stream>>>(
        x, node_anchor, att_w, att_b, node_prompted);

    int tiles   = (N_EDGES + 15) / 16;
    int eblocks = (tiles + 7) / 8;
    edge_kernel<<<eblocks, 256, 0, stream>>>(
        x, edge_anchor, edge_w, edge_b, edge_index, edge_prompt, edge_agg);

    gate_kernel<<<(N_NODES + 255) / 256, 256, 0, stream>>>(
        node_prompted, edge_agg, gate_w, gate_b, final_x);
}